// GatedLaplacianModel_25082609009445
// MI455X (gfx1250) — compile-verified
//
#include <hip/hip_runtime.h>
#include <hip/hip_bf16.h>

#define Nn 4096
#define Dd 128
#define KNN 10

typedef float v2f __attribute__((ext_vector_type(2)));
typedef float v8f __attribute__((ext_vector_type(8)));

// ---------------- kernel 1: gate + masked (written to d_out tail) ----------
__global__ void masked_kernel(const float* __restrict__ x,
                              const float* __restrict__ alpha,
                              const float* __restrict__ noise,
                              float* __restrict__ masked) {
    int idx = blockIdx.x * blockDim.x + threadIdx.x;   // N*D total
    int j = idx & (Dd - 1);
    float z = alpha[j] + 0.5f * noise[j] + 0.5f;
    float g = fminf(fmaxf(z, 0.0f), 1.0f);
    masked[idx] = x[idx] * g;
}

// ---------------- kernel 2: row squared norms ------------------------------
__global__ void rownorm_kernel(const float* __restrict__ masked,
                               float* __restrict__ r) {
    __shared__ float red[128];
    int row = blockIdx.x;
    int t = threadIdx.x;                    // 128 threads
    float v = masked[(size_t)row * Dd + t];
    red[t] = v * v;
    __syncthreads();
    for (int s = 64; s > 0; s >>= 1) {
        if (t < s) red[t] += red[t + s];
        __syncthreads();
    }
    if (t == 0) r[row] = red[0];
}

// ---------------- kernel 3: Dx = r_i - 2*X X^T + r_j via fp32 WMMA ---------
// block = 256 threads (8 waves), computes a 64x64 tile of Dx.
__global__ __launch_bounds__(256) void dist_wmma_kernel(
        const float* __restrict__ Xm, const float* __restrict__ r,
        float* __restrict__ Dx) {
    __shared__ float As[64][36];   // stride 36 floats: conflict-free, 16B aligned
    __shared__ float Bs[64][36];
    __shared__ float rI[64];
    __shared__ float rJ[64];

    const int i0 = blockIdx.y * 64;
    const int j0 = blockIdx.x * 64;
    const int t    = threadIdx.x;
    const int wave = t >> 5;
    const int lane = t & 31;
    const int lr = lane & 15;      // fragment row within 16
    const int hi = lane >> 4;      // 0/1 half selector (K pairs / M+8)
    const int sc  = wave & 3;      // subtile column 0..3
    const int sr0 = wave >> 2;     // subtile rows sr0 and sr0+2

    if (t < 64)       rI[t]      = r[i0 + t];
    else if (t < 128) rJ[t - 64] = r[j0 + (t - 64)];

    v8f acc0 = {};
    v8f acc1 = {};

    const int lrow  = t >> 2;        // 0..63 : row loaded by this thread
    const int lcol8 = (t & 3) * 8;   // 0,8,16,24 : 8 floats per thread

    for (int kb = 0; kb < Dd; kb += 32) {
        __syncthreads();
        {   // stage 64x32 of A-rows and B-rows into LDS (float4 x2 each)
            const float4* ga = (const float4*)(Xm + (size_t)(i0 + lrow) * Dd + kb + lcol8);
            float4 a0 = ga[0], a1 = ga[1];
            float* as = &As[lrow][lcol8];
            ((float4*)as)[0] = a0; ((float4*)as)[1] = a1;

            const float4* gb = (const float4*)(Xm + (size_t)(j0 + lrow) * Dd + kb + lcol8);
            float4 b0 = gb[0], b1 = gb[1];
            float* bs = &Bs[lrow][lcol8];
            ((float4*)bs)[0] = b0; ((float4*)bs)[1] = b1;
        }
        __syncthreads();

#pragma unroll
        for (int k = 0; k < 32; k += 4) {
            // A 16x4 fp32 fragment: v0 = K=2*hi, v1 = K=2*hi+1, row = lr
            v2f a0 = *(const v2f*)&As[sr0 * 16 + lr][k + 2 * hi];
            v2f a1 = *(const v2f*)&As[(sr0 + 2) * 16 + lr][k + 2 * hi];
            // B = Xm^T  ==> identical per-lane load pattern from the column tile
            v2f b  = *(const v2f*)&Bs[sc * 16 + lr][k + 2 * hi];
            acc0 = __builtin_amdgcn_wmma_f32_16x16x4_f32(
                       false, a0, false, b, (short)0, acc0, false, false);
            acc1 = __builtin_amdgcn_wmma_f32_16x16x4_f32(
                       false, a1, false, b, (short)0, acc1, false, false);
        }
    }

    // C/D layout: element (M = v + 8*hi, N = lr) in VGPR v
    const int col = j0 + sc * 16 + lr;
    const float rj = rJ[sc * 16 + lr];
#pragma unroll
    for (int v = 0; v < 8; v++) {
        int row0 = sr0 * 16 + v + 8 * hi;
        int row1 = (sr0 + 2) * 16 + v + 8 * hi;
        Dx[(size_t)(i0 + row0) * Nn + col] = rI[row0] - 2.0f * acc0[v] + rj;
        Dx[(size_t)(i0 + row1) * Nn + col] = rI[row1] - 2.0f * acc1[v] + rj;
    }
}

// ---------------- kernel 4: 10th-smallest per row (exact order statistic) --
__global__ __launch_bounds__(256) void knn_kernel(const float* __restrict__ Dx,
                                                  float* __restrict__ knn) {
    __shared__ float smin[256];
    __shared__ int   scnt[256];
    const int row = blockIdx.x;
    const int t = threadIdx.x;
    const float* rp = Dx + (size_t)row * Nn;
    float vals[16];
#pragma unroll
    for (int i = 0; i < 16; i++) vals[i] = rp[t + 256 * i];

    float thresh = -__builtin_inff();
    float m = 0.0f;
    int cnt = 0;
    while (cnt < KNN) {
        float lm = __builtin_inff();
#pragma unroll
        for (int i = 0; i < 16; i++) {
            float v = vals[i];
            if (v > thresh && v < lm) lm = v;
        }
        smin[t] = lm;
        __syncthreads();
        for (int s = 128; s > 0; s >>= 1) {
            if (t < s) smin[t] = fminf(smin[t], smin[t + s]);
            __syncthreads();
        }
        m = smin[0];
        int lc = 0;
#pragma unroll
        for (int i = 0; i < 16; i++) lc += (vals[i] == m) ? 1 : 0;
        __syncthreads();
        scnt[t] = lc;
        __syncthreads();
        for (int s = 128; s > 0; s >>= 1) {
            if (t < s) scnt[t] += scnt[t + s];
            __syncthreads();
        }
        cnt += scnt[0];
        thresh = m;
        __syncthreads();
    }
    if (t == 0) knn[row] = m;
}

// ---------------- kernel 5: median of knn via bitonic sort -> -1/(2*sigma) -
__global__ __launch_bounds__(1024) void median_kernel(const float* __restrict__ knn,
                                                      float* __restrict__ scalar) {
    __shared__ float s[Nn];
    const int t = threadIdx.x;
    for (int i = t; i < Nn; i += 1024) s[i] = knn[i];
    __syncthreads();
    for (int ksz = 2; ksz <= Nn; ksz <<= 1) {
        for (int j = ksz >> 1; j > 0; j >>= 1) {
            for (int i = t; i < Nn; i += 1024) {
                int l = i ^ j;
                if (l > i) {
                    bool asc = ((i & ksz) == 0);
                    float a = s[i], b = s[l];
                    if ((a > b) == asc) { s[i] = b; s[l] = a; }
                }
            }
            __syncthreads();
        }
    }
    if (t == 0) {
        float sigma = 0.5f * (s[Nn / 2] + s[Nn / 2 - 1]);
        if (sigma < 1e-8f) sigma = 1.0f;
        scalar[0] = -1.0f / (2.0f * sigma);   // -1/(FAC*sigma), FAC=2
    }
}

// ---------------- kernel 6: P = exp(Dx * c) row-normalized (in place) ------
__global__ __launch_bounds__(256) void norm_kernel(float* __restrict__ P,
                                                   const float* __restrict__ scalar) {
    __shared__ float red[256];
    const int row = blockIdx.x;
    const int t = threadIdx.x;
    const float c = scalar[0];
    float* rp = P + (size_t)row * Nn;
    float w[16];
    float sum = 0.0f;
#pragma unroll
    for (int i = 0; i < 16; i++) {
        float e = __expf(rp[t + 256 * i] * c);
        w[i] = e;
        sum += e;
    }
    red[t] = sum;
    __syncthreads();
    for (int s = 128; s > 0; s >>= 1) {
        if (t < s) red[t] += red[t + s];
        __syncthreads();
    }
    float inv = 1.0f / red[0];
#pragma unroll
    for (int i = 0; i < 16; i++) rp[t + 256 * i] = w[i] * inv;
}

// ---------------- launcher -------------------------------------------------
extern "C" void kernel_launch(void* const* d_in, const int* in_sizes, int n_in,
                              void* d_out, int out_size, void* d_ws, size_t ws_size,
                              hipStream_t stream) {
    const float* x     = (const float*)d_in[0];
    const float* alpha = (const float*)d_in[1];
    const float* noise = (const float*)d_in[2];

    float* P      = (float*)d_out;                    // N*N (first tuple output)
    float* masked = P + (size_t)Nn * Nn;              // N*D (second tuple output)

    float* ws     = (float*)d_ws;
    float* r      = ws;                               // N floats
    float* knn    = ws + Nn;                          // N floats
    float* scalar = ws + 2 * Nn;                      // 1 float

    masked_kernel<<<(Nn * Dd) / 256, 256, 0, stream>>>(x, alpha, noise, masked);
    rownorm_kernel<<<Nn, 128, 0, stream>>>(masked, r);
    dim3 grid(Nn / 64, Nn / 64);
    dist_wmma_kernel<<<grid, 256, 0, stream>>>(masked, r, P);
    knn_kernel<<<Nn, 256, 0, stream>>>(P, knn);
    median_kernel<<<1, 1024, 0, stream>>>(knn, scalar);
    norm_kernel<<<Nn, 256, 0, stream>>>(P, scalar);
}